// MLAPModel_10282151707710
// MI455X (gfx1250) — compile-verified
//
#include <hip/hip_runtime.h>
#include <math.h>

// Problem constants (match reference)
#define NN 20000
#define EE 320000
#define DD 128
#define HH 3
#define LL 3
#define GG 64
#define NEG_SLOPE 0.2f

typedef float v2f __attribute__((ext_vector_type(2)));
typedef float v8f __attribute__((ext_vector_type(8)));

// ---------------------------------------------------------------------------
// CAS-based float atomic max (safe across toolchains; init with -inf)
// ---------------------------------------------------------------------------
__device__ __forceinline__ void atomicMaxF(float* addr, float val) {
    unsigned int* ua = (unsigned int*)addr;
    unsigned int old = *ua;
    while (__uint_as_float(old) < val) {
        unsigned int assumed = old;
        old = atomicCAS(ua, assumed, __float_as_uint(val));
        if (old == assumed) break;
    }
}

// ---------------------------------------------------------------------------
// WMMA fp32 GEMM:  C[M x Nc] = A[M x K] * W[Nc x K]^T  (+bias, relu, +resid)
// One wave computes a 16x64 tile = 4 accumulators of V_WMMA_F32_16X16X4_F32.
// The A fragment is loaded once per k-step and reused by 4 consecutive WMMAs
// (1.25 loads per wmma instead of 2).  Nc must be a multiple of 64.
// A-frag layout (32b 16x4): lanes 0-15 hold K=k0,k0+1; lanes 16-31 K=k0+2,k0+3.
// B-frag (4x16 from row-major W[Nc x K]) loads with the identical pattern.
// C/D: VGPR i -> row = i + 8*(lane>>4), col = lane&15.
// ---------------------------------------------------------------------------
__global__ void gemm_wmma_kernel(const float* __restrict__ A,
                                 const float* __restrict__ W,
                                 float* __restrict__ C,
                                 const float* __restrict__ bias,
                                 const float* __restrict__ resid,
                                 int M, int Nc, int K, int relu_flag)
{
    const int lane = threadIdx.x & 31;
    const int hlf  = lane >> 4;        // 0 or 1
    const int r    = lane & 15;
    const int row0 = blockIdx.x * 16;
    const int col0 = (blockIdx.y * blockDim.y + threadIdx.y) * 64;
    if (row0 >= M || col0 >= Nc) return;   // wave-uniform -> EXEC all-1 inside

    const float* Arow = A + (size_t)(row0 + r) * K + 2 * hlf;
    const float* Wrow = W + (size_t)(col0 + r) * K + 2 * hlf;
    const size_t wstep = (size_t)16 * K;   // stride between 16-col tiles of W

    v8f acc[4] = {};
#pragma unroll 2
    for (int k0 = 0; k0 < K; k0 += 4) {
        v2f a  = *(const v2f*)(Arow + k0);
        v2f b0 = *(const v2f*)(Wrow + k0);
        v2f b1 = *(const v2f*)(Wrow + wstep + k0);
        v2f b2 = *(const v2f*)(Wrow + 2 * wstep + k0);
        v2f b3 = *(const v2f*)(Wrow + 3 * wstep + k0);
        acc[0] = __builtin_amdgcn_wmma_f32_16x16x4_f32(false, a, false, b0,
                                                       (short)0, acc[0], false, false);
        acc[1] = __builtin_amdgcn_wmma_f32_16x16x4_f32(false, a, false, b1,
                                                       (short)0, acc[1], false, false);
        acc[2] = __builtin_amdgcn_wmma_f32_16x16x4_f32(false, a, false, b2,
                                                       (short)0, acc[2], false, false);
        acc[3] = __builtin_amdgcn_wmma_f32_16x16x4_f32(false, a, false, b3,
                                                       (short)0, acc[3], false, false);
    }

#pragma unroll
    for (int j = 0; j < 4; ++j) {
        const int col = col0 + j * 16 + r;
#pragma unroll
        for (int i = 0; i < 8; ++i) {
            const int row = row0 + i + 8 * hlf;
            float v = acc[j][i];
            if (bias)      v += bias[col];
            if (relu_flag) v = v > 0.0f ? v : 0.0f;
            if (resid)     v += resid[(size_t)row * Nc + col];
            C[(size_t)row * Nc + col] = v;
        }
    }
}

// ---------------------------------------------------------------------------
// Per-(node,head) attention scores: s_ns = <h, a_ns>, s_nd = <h, a_nd>
// ---------------------------------------------------------------------------
__global__ void scores_kernel(const float* __restrict__ hbuf,
                              const float* __restrict__ a_ns,
                              const float* __restrict__ a_nd,
                              float* __restrict__ s_ns,
                              float* __restrict__ s_nd)
{
    int t = blockIdx.x * blockDim.x + threadIdx.x;
    if (t >= NN * HH) return;
    int n = t / HH, h = t % HH;
    const float4* hv = (const float4*)(hbuf + (size_t)n * HH * DD + h * DD);
    const float4* wp = (const float4*)(a_ns + h * DD);
    const float4* wq = (const float4*)(a_nd + h * DD);
    float s1 = 0.0f, s2 = 0.0f;
#pragma unroll 8
    for (int i = 0; i < DD / 4; ++i) {
        float4 x = hv[i], p = wp[i], q = wq[i];
        s1 += x.x * p.x + x.y * p.y + x.z * p.z + x.w * p.w;
        s2 += x.x * q.x + x.y * q.y + x.z * q.z + x.w * q.w;
    }
    s_ns[t] = s1;
    s_nd[t] = s2;
}

__global__ void fill_kernel(float* __restrict__ p, float v, int n)
{
    int i = blockIdx.x * blockDim.x + threadIdx.x;
    if (i < n) p[i] = v;
}

// edge logit = leaky_relu(s_ns[src] + s_nd[dst]); segment max over dst
__global__ void edge_max_kernel(const int* __restrict__ src,
                                const int* __restrict__ dst,
                                const float* __restrict__ s_ns,
                                const float* __restrict__ s_nd,
                                float* __restrict__ elog,
                                float* __restrict__ seg_m)
{
    int e = blockIdx.x * blockDim.x + threadIdx.x;
    if (e >= EE) return;
    int s = src[e], d = dst[e];
#pragma unroll
    for (int h = 0; h < HH; ++h) {
        float x = s_ns[s * HH + h] + s_nd[d * HH + h];
        x = x > 0.0f ? x : NEG_SLOPE * x;          // leaky relu
        elog[e * HH + h] = x;
        atomicMaxF(&seg_m[d * HH + h], x);
    }
}

// e = exp(logit - max); segment sum over dst
__global__ void edge_expsum_kernel(const int* __restrict__ dst,
                                   float* __restrict__ elog,
                                   const float* __restrict__ seg_m,
                                   float* __restrict__ seg_s)
{
    int e = blockIdx.x * blockDim.x + threadIdx.x;
    if (e >= EE) return;
    int d = dst[e];
#pragma unroll
    for (int h = 0; h < HH; ++h) {
        float v = expf(elog[e * HH + h] - seg_m[d * HH + h]);
        elog[e * HH + h] = v;
        atomicAdd(&seg_s[d * HH + h], v);
    }
}

// msg = h[src] * alpha; agg = segment_sum over dst.  One wave per edge.
__global__ void edge_agg_kernel(const int* __restrict__ src,
                                const int* __restrict__ dst,
                                const float* __restrict__ elog,
                                const float* __restrict__ seg_s,
                                const float* __restrict__ hbuf,
                                float* __restrict__ agg)
{
    int e    = (blockIdx.x * blockDim.x + threadIdx.x) >> 5;
    int lane = threadIdx.x & 31;
    if (e >= EE) return;
    int s = src[e], d = dst[e];
#pragma unroll
    for (int h = 0; h < HH; ++h) {
        float w = elog[e * HH + h] / seg_s[d * HH + h];
        float4 hv = *(const float4*)(hbuf + (size_t)s * HH * DD + h * DD + lane * 4);
        float* o = agg + (size_t)d * HH * DD + h * DD + lane * 4;
        atomicAdd(o + 0, w * hv.x);
        atomicAdd(o + 1, w * hv.y);
        atomicAdd(o + 2, w * hv.z);
        atomicAdd(o + 3, w * hv.w);
    }
}

// gate[n] = <gate1[n,:], p2_w> + p2_b  (one wave per node, wave32 reduce)
__global__ void gate_reduce_kernel(const float* __restrict__ gate1,
                                   const float* __restrict__ p2_w,
                                   const float* __restrict__ p2_b,
                                   float* __restrict__ gate)
{
    int n    = (blockIdx.x * blockDim.x + threadIdx.x) >> 5;
    int lane = threadIdx.x & 31;
    if (n >= NN) return;
    float s = 0.0f;
#pragma unroll
    for (int j = lane; j < 2 * DD; j += 32)
        s += gate1[(size_t)n * 2 * DD + j] * p2_w[j];
#pragma unroll
    for (int off = 16; off > 0; off >>= 1)
        s += __shfl_down(s, off, 32);
    if (lane == 0) gate[n] = s + p2_b[0];
}

__global__ void pool_max_kernel(const float* __restrict__ gate,
                                const int* __restrict__ gid,
                                float* __restrict__ pool_m)
{
    int n = blockIdx.x * blockDim.x + threadIdx.x;
    if (n >= NN) return;
    atomicMaxF(&pool_m[gid[n]], gate[n]);
}

__global__ void pool_expsum_kernel(float* __restrict__ gate,
                                   const int* __restrict__ gid,
                                   const float* __restrict__ pool_m,
                                   float* __restrict__ pool_s)
{
    int n = blockIdx.x * blockDim.x + threadIdx.x;
    if (n >= NN) return;
    int g = gid[n];
    float v = expf(gate[n] - pool_m[g]);
    gate[n] = v;
    atomicAdd(&pool_s[g], v);
}

// out[g,:] += (alpha / L) * feat[n,:]   (one wave per node; folds layer mean)
__global__ void pool_acc_kernel(const float* __restrict__ gate,
                                const float* __restrict__ pool_s,
                                const int* __restrict__ gid,
                                const float* __restrict__ feat,
                                float* __restrict__ out)
{
    int n    = (blockIdx.x * blockDim.x + threadIdx.x) >> 5;
    int lane = threadIdx.x & 31;
    if (n >= NN) return;
    int g = gid[n];
    float alpha = gate[n] / pool_s[g] * (1.0f / (float)LL);
    float4 f = *(const float4*)(feat + (size_t)n * DD + lane * 4);
    float* o = out + (size_t)g * DD + lane * 4;
    atomicAdd(o + 0, alpha * f.x);
    atomicAdd(o + 1, alpha * f.y);
    atomicAdd(o + 2, alpha * f.z);
    atomicAdd(o + 3, alpha * f.w);
}

// ---------------------------------------------------------------------------
// Host orchestration
// ---------------------------------------------------------------------------
extern "C" void kernel_launch(void* const* d_in, const int* in_sizes, int n_in,
                              void* d_out, int out_size, void* d_ws, size_t ws_size,
                              hipStream_t stream)
{
    const float* feat0   = (const float*)d_in[0];   // [N, D]
    const int*   src     = (const int*)  d_in[1];   // [E]
    const int*   dst     = (const int*)  d_in[2];   // [E]
    const int*   gid     = (const int*)  d_in[3];   // [N]
    const float* fc_w    = (const float*)d_in[4];   // [L, H*D, D]
    const float* a_ns    = (const float*)d_in[5];   // [L, H, D]
    const float* a_nd    = (const float*)d_in[6];   // [L, H, D]
    const float* trans_w = (const float*)d_in[7];   // [L, D, H*D]
    const float* trans_b = (const float*)d_in[8];   // [L, D]
    const float* p1_w    = (const float*)d_in[9];   // [L, 2D, D]
    const float* p1_b    = (const float*)d_in[10];  // [L, 2D]
    const float* p2_w    = (const float*)d_in[11];  // [L, 1, 2D]
    const float* p2_b    = (const float*)d_in[12];  // [L, 1]
    float* out = (float*)d_out;                     // [G, D]

    // Workspace layout (floats)
    const size_t ND  = (size_t)NN * DD;          // 2,560,000
    const size_t NHD = (size_t)NN * HH * DD;     // 7,680,000
    const size_t NH  = (size_t)NN * HH;          //    60,000
    const size_t EH  = (size_t)EE * HH;          //   960,000

    float* ws    = (float*)d_ws;
    float* featA = ws;
    float* featB = featA + ND;
    float* hbuf  = featB + ND;          // [N, H*D]; reused as gate1 [N, 2D]
    float* s_ns  = hbuf  + NHD;
    float* s_nd  = s_ns  + NH;
    float* seg_m = s_nd  + NH;
    float* seg_s = seg_m + NH;
    float* elog  = seg_s + NH;          // [E, H]
    float* agg   = elog  + EH;          // [N, H*D]
    float* gate  = agg   + NHD;         // [N]
    float* pool_m = gate + NN;          // [G]
    float* pool_s = pool_m + GG;        // [G]
    float* gate1 = hbuf;                // alias (hbuf dead after aggregation)

    const int TB = 256;
    const dim3 gblock(32, 2);           // 2 waves per block; 1 wave = 16x64 tile

    // output accumulator (harness poisons d_out)
    fill_kernel<<<(GG * DD + TB - 1) / TB, TB, 0, stream>>>(out, 0.0f, GG * DD);

    float* bufs[2] = { featA, featB };
    const float* cur = feat0;

    for (int l = 0; l < LL; ++l) {
        float* nxt = bufs[l & 1];

        // 1) h = feat @ fc_w[l]^T : M=N, Nc=H*D=384, K=D=128  (6 col groups)
        gemm_wmma_kernel<<<dim3(NN / 16, 3), gblock, 0, stream>>>(
            cur, fc_w + (size_t)l * HH * DD * DD, hbuf,
            nullptr, nullptr, NN, HH * DD, DD, 0);

        // 2) attention scores
        scores_kernel<<<(NN * HH + TB - 1) / TB, TB, 0, stream>>>(
            hbuf, a_ns + (size_t)l * HH * DD, a_nd + (size_t)l * HH * DD, s_ns, s_nd);

        // 3) segment softmax over dst
        fill_kernel<<<((int)NH + TB - 1) / TB, TB, 0, stream>>>(seg_m, -INFINITY, (int)NH);
        fill_kernel<<<((int)NH + TB - 1) / TB, TB, 0, stream>>>(seg_s, 0.0f, (int)NH);
        edge_max_kernel<<<(EE + TB - 1) / TB, TB, 0, stream>>>(src, dst, s_ns, s_nd, elog, seg_m);
        edge_expsum_kernel<<<(EE + TB - 1) / TB, TB, 0, stream>>>(dst, elog, seg_m, seg_s);

        // 4) message aggregation (scatter-add), one wave per edge
        fill_kernel<<<((int)NHD + TB - 1) / TB, TB, 0, stream>>>(agg, 0.0f, (int)NHD);
        edge_agg_kernel<<<(EE * 32 + TB - 1) / TB, TB, 0, stream>>>(src, dst, elog, seg_s, hbuf, agg);

        // 5) out = agg @ trans_w[l]^T + b; relu if l<L-1; + residual (2 col groups)
        gemm_wmma_kernel<<<dim3(NN / 16, 1), gblock, 0, stream>>>(
            agg, trans_w + (size_t)l * DD * HH * DD, nxt,
            trans_b + (size_t)l * DD, cur, NN, DD, HH * DD, (l < LL - 1) ? 1 : 0);

        // 6) attention pooling: gate1 = relu(feat @ p1_w^T + p1_b), Nc=2D=256 (4 groups)
        gemm_wmma_kernel<<<dim3(NN / 16, 2), gblock, 0, stream>>>(
            nxt, p1_w + (size_t)l * 2 * DD * DD, gate1,
            p1_b + (size_t)l * 2 * DD, nullptr, NN, 2 * DD, DD, 1);
        gate_reduce_kernel<<<(NN * 32 + TB - 1) / TB, TB, 0, stream>>>(
            gate1, p2_w + (size_t)l * 2 * DD, p2_b + l, gate);

        fill_kernel<<<1, GG, 0, stream>>>(pool_m, -INFINITY, GG);
        fill_kernel<<<1, GG, 0, stream>>>(pool_s, 0.0f, GG);
        pool_max_kernel<<<(NN + TB - 1) / TB, TB, 0, stream>>>(gate, gid, pool_m);
        pool_expsum_kernel<<<(NN + TB - 1) / TB, TB, 0, stream>>>(gate, gid, pool_m, pool_s);
        pool_acc_kernel<<<(NN * 32 + TB - 1) / TB, TB, 0, stream>>>(gate, pool_s, gid, nxt, out);

        cur = nxt;   // next layer reads this layer's output (residual source too)
    }
}